// PinSAGEModel_24232205484578
// MI455X (gfx1250) — compile-verified
//
#include <hip/hip_runtime.h>
#include <hip/hip_bf16.h>

// ---- problem constants (match reference) ----
#define N_TOTAL 1000000
#define N0 495616
#define N1 45056
#define N2 4096
#define E0 450560
#define E1 40960
#define DD 128
#define PP 4096

typedef float v2f __attribute__((ext_vector_type(2)));
typedef float v8f __attribute__((ext_vector_type(8)));

// ------------------------------------------------------------------
// zero the aggregation accumulators (neigh_sum0, deg0, neigh_sum1, deg1)
// ------------------------------------------------------------------
__global__ void zero_kernel(float* __restrict__ p, long long n) {
    long long stride = (long long)gridDim.x * blockDim.x;
    for (long long i = (long long)blockIdx.x * blockDim.x + threadIdx.x; i < n; i += stride)
        p[i] = 0.0f;
}

// ------------------------------------------------------------------
// layer-0 edge scatter: msg = node_features[input_nodes[esrc[e]]]
// neigh_sum0[edst[e]] += msg ; deg0[edst[e]] += 1
// 128 consecutive threads handle one edge (coalesced 512B row reads)
// ------------------------------------------------------------------
__global__ __launch_bounds__(256) void scatter0_kernel(
    const float* __restrict__ nf, const int* __restrict__ input_nodes,
    const int* __restrict__ esrc, const int* __restrict__ edst,
    float* __restrict__ nsum, float* __restrict__ deg) {
    int g = blockIdx.x * 256 + threadIdx.x;
    int e = g >> 7;
    int d = g & 127;
    if (e >= E0) return;
    int dst = edst[e];
    long long row = (long long)input_nodes[esrc[e]];
    atomicAdd(&nsum[(long long)dst * DD + d], nf[row * DD + d]);
    if (d == 0) atomicAdd(&deg[dst], 1.0f);
}

// ------------------------------------------------------------------
// layer-1 edge scatter: msg = h1[esrc[e]] (no indirection)
// ------------------------------------------------------------------
__global__ __launch_bounds__(256) void scatter1_kernel(
    const float* __restrict__ h1,
    const int* __restrict__ esrc, const int* __restrict__ edst,
    float* __restrict__ nsum, float* __restrict__ deg) {
    int g = blockIdx.x * 256 + threadIdx.x;
    int e = g >> 7;
    int d = g & 127;
    if (e >= E1) return;
    int dst = edst[e];
    long long row = (long long)esrc[e];
    atomicAdd(&nsum[(long long)dst * DD + d], h1[row * DD + d]);
    if (d == 0) atomicAdd(&deg[dst], 1.0f);
}

// ------------------------------------------------------------------
// Fused SAGE dense layer on WMMA f32 16x16x4:
//   Hout[i,:] = act( Aself[i,:] @ Wself + (nsum[i,:]/max(deg[i],1)) @ Wneigh + b )
// One wave computes a 16-row x 128-col tile. 8 waves / block.
// gather_idx != nullptr => Aself rows are node_features[gather_idx[i]].
// ------------------------------------------------------------------
__global__ __launch_bounds__(256) void sage_gemm_kernel(
    const float* __restrict__ Aself, const int* __restrict__ gather_idx,
    const float* __restrict__ nsum, const float* __restrict__ deg,
    const float* __restrict__ Wself, const float* __restrict__ Wneigh,
    const float* __restrict__ bias, float* __restrict__ Hout,
    int ntiles, int do_relu) {
    int lane = threadIdx.x & 31;
    int wave = threadIdx.x >> 5;
    int tile = blockIdx.x * 8 + wave;
    if (tile >= ntiles) return;           // wave-uniform exit (EXEC stays all-ones for WMMA)

    int r  = lane & 15;                   // A row / B col / C col within tile
    int hi = lane >> 4;                   // half-wave select (K offset +2 / M offset +8)
    long long row = (long long)tile * 16 + r;

    long long srow = gather_idx ? (long long)gather_idx[row] : row;
    const float* aself  = Aself + srow * DD;
    const float* aneigh = nsum + row * DD;
    float scale = 1.0f / fmaxf(deg[row], 1.0f);

    v8f acc[8];
    #pragma unroll
    for (int n = 0; n < 8; ++n) acc[n] = (v8f){0,0,0,0,0,0,0,0};

    for (int ks = 0; ks < 32; ++ks) {     // K = 128 in steps of 4
        int kk = ks * 4 + 2 * hi;         // this lane's K base (vgpr v -> K = kk+v)
        v2f a_s = *(const v2f*)(aself + kk);
        v2f a_n;
        a_n.x = aneigh[kk]     * scale;
        a_n.y = aneigh[kk + 1] * scale;
        #pragma unroll
        for (int n = 0; n < 8; ++n) {
            int col = n * 16 + r;
            v2f b_s, b_n;
            b_s.x = Wself[(long long)kk * DD + col];
            b_s.y = Wself[(long long)(kk + 1) * DD + col];
            b_n.x = Wneigh[(long long)kk * DD + col];
            b_n.y = Wneigh[(long long)(kk + 1) * DD + col];
            acc[n] = __builtin_amdgcn_wmma_f32_16x16x4_f32(
                false, a_s, false, b_s, (short)0, acc[n], false, false);
            acc[n] = __builtin_amdgcn_wmma_f32_16x16x4_f32(
                false, a_n, false, b_n, (short)0, acc[n], false, false);
        }
    }

    // epilogue: bias (+ReLU) and store. C/D layout: vgpr rr -> row rr + 8*hi, col = lane&15.
    #pragma unroll
    for (int n = 0; n < 8; ++n) {
        int col = n * 16 + r;
        float bv = bias[col];
        #pragma unroll
        for (int rr = 0; rr < 8; ++rr) {
            float v = acc[n][rr] + bv;
            if (do_relu) v = fmaxf(v, 0.0f);
            Hout[((long long)tile * 16 + rr + 8 * hi) * DD + col] = v;
        }
    }
}

// ------------------------------------------------------------------
// final output gathers: out = concat(h2[pos_s], h2[pos_d], h2[neg_s], h2[neg_d])
// ------------------------------------------------------------------
__global__ __launch_bounds__(256) void gather_out_kernel(
    const float* __restrict__ h2,
    const int* __restrict__ ps, const int* __restrict__ pd,
    const int* __restrict__ ns, const int* __restrict__ nd,
    float* __restrict__ out) {
    int g = blockIdx.x * 256 + threadIdx.x;   // total 4*P*D = 2,097,152
    int seg = g / (PP * DD);
    int rem = g - seg * (PP * DD);
    int i = rem >> 7;
    int d = rem & 127;
    const int* idx = (seg == 0) ? ps : (seg == 1) ? pd : (seg == 2) ? ns : nd;
    out[g] = h2[(long long)idx[i] * DD + d];
}

extern "C" void kernel_launch(void* const* d_in, const int* in_sizes, int n_in,
                              void* d_out, int out_size, void* d_ws, size_t ws_size,
                              hipStream_t stream) {
    const float* nf          = (const float*)d_in[0];
    const int*   input_nodes = (const int*)d_in[1];
    const int*   esrc0       = (const int*)d_in[2];
    const int*   edst0       = (const int*)d_in[3];
    const int*   esrc1       = (const int*)d_in[4];
    const int*   edst1       = (const int*)d_in[5];
    const int*   pos_s       = (const int*)d_in[6];
    const int*   pos_d       = (const int*)d_in[7];
    const int*   neg_s       = (const int*)d_in[8];
    const int*   neg_d       = (const int*)d_in[9];
    const float* Wself0      = (const float*)d_in[10];
    const float* Wneigh0     = (const float*)d_in[11];
    const float* b0          = (const float*)d_in[12];
    const float* Wself1      = (const float*)d_in[13];
    const float* Wneigh1     = (const float*)d_in[14];
    const float* b1          = (const float*)d_in[15];
    float* out = (float*)d_out;

    // workspace layout (floats)
    float* ws = (float*)d_ws;
    long long off = 0;
    float* nsum0 = ws + off; off += (long long)N1 * DD;   // 5,767,168
    float* deg0  = ws + off; off += N1;                   //    45,056
    float* nsum1 = ws + off; off += (long long)N2 * DD;   //   524,288
    float* deg1  = ws + off; off += N2;                   //     4,096
    long long zero_total = off;                           // contiguous accumulator region
    float* h1 = ws + off; off += (long long)N1 * DD;
    float* h2 = ws + off; off += (long long)N2 * DD;      // total ~48.2 MB

    // 1) zero accumulators
    zero_kernel<<<2048, 256, 0, stream>>>(ws, zero_total);

    // 2) layer-0 scatter (double-indirect gather of node features)
    scatter0_kernel<<<(E0 * DD) / 256, 256, 0, stream>>>(
        nf, input_nodes, esrc0, edst0, nsum0, deg0);

    // 3) layer-0 fused WMMA GEMM + bias + ReLU  (45056 rows -> 2816 tiles)
    {
        int ntiles = N1 / 16;
        sage_gemm_kernel<<<(ntiles + 7) / 8, 256, 0, stream>>>(
            nf, input_nodes, nsum0, deg0, Wself0, Wneigh0, b0, h1, ntiles, 1);
    }

    // 4) layer-1 scatter
    scatter1_kernel<<<(E1 * DD) / 256, 256, 0, stream>>>(
        h1, esrc1, edst1, nsum1, deg1);

    // 5) layer-1 fused WMMA GEMM + bias (no relu)  (4096 rows -> 256 tiles)
    {
        int ntiles = N2 / 16;
        sage_gemm_kernel<<<(ntiles + 7) / 8, 256, 0, stream>>>(
            h1, nullptr, nsum1, deg1, Wself1, Wneigh1, b1, h2, ntiles, 0);
    }

    // 6) output gathers (4 * P * D elements)
    gather_out_kernel<<<(4 * PP * DD) / 256, 256, 0, stream>>>(
        h2, pos_s, pos_d, neg_s, neg_d, out);
}